// MeanAddCelltype_7842610282625
// MI455X (gfx1250) — compile-verified
//
#include <hip/hip_runtime.h>
#include <hip/hip_bf16.h>

typedef __attribute__((ext_vector_type(2))) float v2f;
typedef __attribute__((ext_vector_type(8))) float v8f;

#define NROWS 8192
#define DIN   512
#define DHID  1024
#define DOUT  512
#define KNB   16

// ---------------------------------------------------------------------------
// Kernel 1: per-row mask scan -> 16 neighbor indices -> mean + residual.
// One 256-thread block per row. Mask row = 8192 bytes = 32 bytes/thread,
// read as two b128 loads. Index order is irrelevant (sum is permutation-
// invariant), so an LDS atomic-append slot list is exact.
// ---------------------------------------------------------------------------
__global__ __launch_bounds__(256) void gather_mean_residual(
    const float* __restrict__ x, const unsigned char* __restrict__ mask,
    float* __restrict__ hidden)
{
    __shared__ int s_cnt;
    __shared__ int s_idx[KNB];
    const int row = blockIdx.x;
    const int t   = threadIdx.x;
    if (t == 0) s_cnt = 0;
    __syncthreads();

    const uint4* mrow = (const uint4*)(mask + (size_t)row * NROWS);
    uint4 m0 = mrow[t * 2 + 0];
    uint4 m1 = mrow[t * 2 + 1];
    unsigned int words[8] = {m0.x, m0.y, m0.z, m0.w, m1.x, m1.y, m1.z, m1.w};
    const int base = t * 32;
#pragma unroll
    for (int w = 0; w < 8; ++w) {
        unsigned int v = words[w];
        if (v) {
#pragma unroll
            for (int b = 0; b < 4; ++b) {
                if ((v >> (8 * b)) & 0xFFu) {
                    int slot = atomicAdd(&s_cnt, 1);
                    if (slot < KNB) s_idx[slot] = base + w * 4 + b;
                }
            }
        }
    }
    __syncthreads();

    int nidx[KNB];
#pragma unroll
    for (int j = 0; j < KNB; ++j) nidx[j] = s_idx[j];

    // 512 dims / 256 threads = 2 dims per thread
#pragma unroll
    for (int rep = 0; rep < DIN / 256; ++rep) {
        const int d = t + rep * 256;
        float acc = 0.0f;
#pragma unroll
        for (int j = 0; j < KNB; ++j)
            acc += x[(size_t)nidx[j] * DIN + d];
        hidden[(size_t)row * DIN + d] =
            x[(size_t)row * DIN + d] + acc * (1.0f / (float)KNB);
    }
}

// ---------------------------------------------------------------------------
// Weight pre-pack: Bp[k2*N + n] = { B[2*k2][n], B[2*k2+1][n] }.
// Makes each WMMA B-fragment a single b64 load in the GEMM hot loop
// (instead of two b32 loads N*4 bytes apart). One-shot, 2 MB per weight.
// ---------------------------------------------------------------------------
template <int K, int N>
__global__ __launch_bounds__(256) void pack_b(const float* __restrict__ B,
                                              v2f* __restrict__ Bp)
{
    const int idx = blockIdx.x * 256 + threadIdx.x;   // over (K/2)*N
    const int k2  = idx / N;
    const int n   = idx % N;
    v2f p;
    p.x = B[(size_t)(2 * k2 + 0) * N + n];
    p.y = B[(size_t)(2 * k2 + 1) * N + n];
    Bp[idx] = p;
}

// ---------------------------------------------------------------------------
// Kernels 2/3: fp32 WMMA GEMM, C = act(A[MxK] * Bpacked + bias).
// One wave owns a 32x64 output tile: 8 accumulators (v8f), K stepped by 4
// with V_WMMA_F32_16X16X4_F32, software-pipelined. 6 vmem (all b64) per
// 8 WMMAs. K/N compile-time so all strides fold into immediate offsets.
// Fragment layouts per CDNA5 ISA 7.12.2:
//   A 16x4 (2 VGPR):  m = lane&15,  k = vgpr + 2*(lane>>4)
//   B 4x16 (2 VGPR):  n = lane&15,  k = vgpr + 2*(lane>>4)
//   C/D 16x16 (8 VGPR): m = r + 8*(lane>>4), n = lane&15
// ---------------------------------------------------------------------------
#define WMMA_F32(a, b, c) \
    __builtin_amdgcn_wmma_f32_16x16x4_f32(false, (a), false, (b), (short)0, (c), false, false)

template <int K, int N, int RELU>
__global__ __launch_bounds__(256) void gemm_bias(
    const float* __restrict__ A, const v2f* __restrict__ Bp,
    const float* __restrict__ bias, float* __restrict__ C, int M)
{
    const int lane = threadIdx.x & 31;
    const int wid  = (blockIdx.x * blockDim.x + threadIdx.x) >> 5;
    constexpr int nStrips = N / 64;          // 64 cols per wave
    const int rowTile = wid / nStrips;       // 32 rows per wave
    const int strip   = wid % nStrips;
    if (rowTile * 32 >= M) return;           // wave-uniform (EXEC stays all-1s)

    const int mr = lane & 15;
    const int kq = lane >> 4;                // k half-select (0 or 1 k-pairs)
    const int m0 = rowTile * 32 + mr;

    const float* Ap0 = A + (size_t)m0 * K + kq * 2;
    const float* Ap1 = Ap0 + (size_t)16 * K;
    // packed B: fragment j of k-step kk lives at Bw[(kk/2)*N + j*16]
    const v2f* Bw = Bp + (size_t)kq * N + strip * 64 + mr;

    v8f acc[8] = {v8f{}, v8f{}, v8f{}, v8f{}, v8f{}, v8f{}, v8f{}, v8f{}};

    // prologue: fragments for k-step 0
    v2f a0, a1, b[4];
    a0.x = Ap0[0]; a0.y = Ap0[1];
    a1.x = Ap1[0]; a1.y = Ap1[1];
#pragma unroll
    for (int j = 0; j < 4; ++j) b[j] = Bw[j * 16];

#pragma unroll 2
    for (int kk = 0; kk < K - 4; kk += 4) {
        // prefetch next k-step (b64 immediate-offset loads)
        v2f a0n, a1n, bn[4];
        a0n.x = Ap0[kk + 4]; a0n.y = Ap0[kk + 5];
        a1n.x = Ap1[kk + 4]; a1n.y = Ap1[kk + 5];
        const v2f* Bn = Bw + (size_t)((kk >> 1) + 2) * N;
#pragma unroll
        for (int j = 0; j < 4; ++j) bn[j] = Bn[j * 16];

        // consume current fragments: 8 WMMAs, B reused across 2 row-tiles
#pragma unroll
        for (int j = 0; j < 4; ++j) {
            acc[j]     = WMMA_F32(a0, b[j], acc[j]);
            acc[j + 4] = WMMA_F32(a1, b[j], acc[j + 4]);
        }
        a0 = a0n; a1 = a1n;
#pragma unroll
        for (int j = 0; j < 4; ++j) b[j] = bn[j];
    }
    // final k-step
#pragma unroll
    for (int j = 0; j < 4; ++j) {
        acc[j]     = WMMA_F32(a0, b[j], acc[j]);
        acc[j + 4] = WMMA_F32(a1, b[j], acc[j + 4]);
    }

    // epilogue: bias (+ReLU), coalesced 16-lane row segments
    const int rbase = kq << 3;               // 0 or 8
#pragma unroll
    for (int half = 0; half < 2; ++half) {
#pragma unroll
        for (int j = 0; j < 4; ++j) {
            const int n    = strip * 64 + j * 16 + mr;
            const float bv = bias[n];
#pragma unroll
            for (int r = 0; r < 8; ++r) {
                const int mm = rowTile * 32 + half * 16 + rbase + r;
                float v = acc[half * 4 + j][r] + bv;
                if (RELU) v = v > 0.0f ? v : 0.0f;
                C[(size_t)mm * N + n] = v;
            }
        }
    }
}

// ---------------------------------------------------------------------------
// Launch: inputs (setup_inputs order):
//   0:x [8192*512 f32] 1:real_edge_mask [bool, unused] 2:fake_edge_mask [bool]
//   3:W1 [512*1024]    4:b1 [1024]      5:W2 [1024*512] 6:b2 [512]
// Workspace: hidden 16MB | h 32MB | W1 packed 2MB | W2 packed 2MB  (52MB)
// ---------------------------------------------------------------------------
extern "C" void kernel_launch(void* const* d_in, const int* in_sizes, int n_in,
                              void* d_out, int out_size, void* d_ws, size_t ws_size,
                              hipStream_t stream) {
    const float*         x    = (const float*)d_in[0];
    const unsigned char* fmsk = (const unsigned char*)d_in[2];
    const float*         W1   = (const float*)d_in[3];
    const float*         b1   = (const float*)d_in[4];
    const float*         W2   = (const float*)d_in[5];
    const float*         b2   = (const float*)d_in[6];
    float*               out  = (float*)d_out;

    float* hidden = (float*)d_ws;                           // [8192,512]
    float* h      = hidden + (size_t)NROWS * DIN;           // [8192,1024]
    v2f*   W1p    = (v2f*)(h + (size_t)NROWS * DHID);       // [(512/2)*1024]
    v2f*   W2p    = W1p + (size_t)(DIN / 2) * DHID;         // [(1024/2)*512]

    // 0) pack weights for b64 fragment loads (one-shot, 2 MB each)
    pack_b<DIN,  DHID><<<(DIN / 2) * DHID / 256, 256, 0, stream>>>(W1, W1p);
    pack_b<DHID, DOUT><<<(DHID / 2) * DOUT / 256, 256, 0, stream>>>(W2, W2p);

    // 1) neighbor mean + residual
    gather_mean_residual<<<NROWS, 256, 0, stream>>>(x, fmsk, hidden);

    // 2) h = relu(hidden @ W1 + b1): (8192/32)*(1024/64) = 4096 waves -> 512 blocks
    {
        const int waves  = (NROWS / 32) * (DHID / 64);
        const int blocks = waves * 32 / 256;
        gemm_bias<DIN, DHID, 1><<<blocks, 256, 0, stream>>>(hidden, W1p, b1, h, NROWS);
    }

    // 3) out = h @ W2 + b2: (8192/32)*(512/64) = 2048 waves -> 256 blocks
    {
        const int waves  = (NROWS / 32) * (DOUT / 64);
        const int blocks = waves * 32 / 256;
        gemm_bias<DHID, DOUT, 0><<<blocks, 256, 0, stream>>>(h, W2p, b2, out, NROWS);
    }
}